// LuongLocalAttention_4664334483965
// MI455X (gfx1250) — compile-verified
//
#include <hip/hip_runtime.h>
#include <math.h>

#define B_  32
#define S_  4096
#define D_  1024
#define U_  1024

#define NCHUNK 32
#define ROWS_PER_BLOCK (S_ / NCHUNK)     // 128 key rows per block
#define GROUP 8                          // rows processed per inner step (== #waves)
#define NGROUPS (ROWS_PER_BLOCK / GROUP) // 16

typedef float v2f __attribute__((ext_vector_type(2)));
typedef float v8f __attribute__((ext_vector_type(8)));

__device__ __forceinline__ float waveReduceSum(float v) {
    #pragma unroll
    for (int off = 16; off > 0; off >>= 1)
        v += __shfl_xor(v, off, 32);
    return v;
}

// ---------------------------------------------------------------------------
// Kernel 1: act = tanh(query @ Wq), exact f32 GEMM via V_WMMA_F32_16X16X4_F32.
// One wave per 16x16 output tile. grid = (U/16, B/16), block = 32 (one wave,
// EXEC all-ones as WMMA requires).
//
// A (16x4 f32) layout: lane L holds M = L&15; VGPR i holds K = 2*(L>>4) + i.
// B (4x16 f32) layout (mirrored): lane L holds N = L&15; VGPR i holds
//   K = 2*(L>>4) + i.
// C/D (16x16 f32): lane L holds N = L&15; VGPR v holds M = v + 8*(L>>4).
// ---------------------------------------------------------------------------
__global__ __launch_bounds__(32)
void wq_gemm_tanh_kernel(const float* __restrict__ query,
                         const float* __restrict__ Wq,
                         float* __restrict__ act) {
    const int lane = threadIdx.x;
    const int m0   = blockIdx.y * 16;
    const int n0   = blockIdx.x * 16;
    const int low  = lane & 15;
    const int half = lane >> 4;          // 0 or 1

    const int mrow = m0 + low;
    const int ncol = n0 + low;

    v8f c = {};
    for (int k0 = 0; k0 < D_; k0 += 4) {
        const int ka = k0 + 2 * half;
        v2f a, b;
        a.x = query[mrow * D_ + ka];
        a.y = query[mrow * D_ + ka + 1];
        b.x = Wq[(size_t)ka       * U_ + ncol];
        b.y = Wq[(size_t)(ka + 1) * U_ + ncol];
        c = __builtin_amdgcn_wmma_f32_16x16x4_f32(false, a, false, b,
                                                  (short)0, c, false, false);
    }

    #pragma unroll
    for (int v = 0; v < 8; ++v) {
        const int mr = m0 + v + 8 * half;
        act[mr * U_ + ncol] = tanhf(c[v]);
    }
}

// ---------------------------------------------------------------------------
// Kernel 2: per-batch gauss scalar.
//   sig  = sigmoid(sum_u act[b,u] * wp[u])
//   g[b] = exp( -(kl*(1-sig))^2 / (2*(win/2)^2) )
// grid = B, block = 256.
// ---------------------------------------------------------------------------
__global__ __launch_bounds__(256)
void gauss_kernel(const float* __restrict__ act,
                  const float* __restrict__ wp,
                  const float* __restrict__ key_lengths,
                  const int*   __restrict__ window_size,
                  float* __restrict__ g_out) {
    __shared__ float warpsum[8];
    const int b = blockIdx.x, t = threadIdx.x;
    float p = 0.f;
    for (int u = t; u < U_; u += 256) p += act[b * U_ + u] * wp[u];
    p = waveReduceSum(p);
    if ((t & 31) == 0) warpsum[t >> 5] = p;
    __syncthreads();
    if (t == 0) {
        float tot = 0.f;
        #pragma unroll
        for (int i = 0; i < 8; ++i) tot += warpsum[i];
        const float sig  = 1.0f / (1.0f + expf(-tot));
        const float kl   = key_lengths[b];
        const float win  = (float)window_size[0];
        const float std2 = (win * 0.5f) * (win * 0.5f);
        const float pa   = kl * sig;
        const float dist = (kl - pa) * (kl - pa);
        g_out[b] = expf(-dist / (2.0f * std2));
    }
}

// ---------------------------------------------------------------------------
// Kernel 3: fused score + online-softmax + context pass. Reads keys from HBM
// exactly once. Each thread owns d-slice [4t, 4t+4); a block covers one full
// key row per float4 load group (256 * 16B = 4KB, perfectly coalesced).
// grid = (NCHUNK, B), block = 256 (8 waves). Writes raw scores to align_out
// and per-chunk (m, l, ctx[D]) partials to workspace.
// ---------------------------------------------------------------------------
__global__ __launch_bounds__(256)
void flash_pass_kernel(const float* __restrict__ query,
                       const float* __restrict__ keys,
                       float* __restrict__ align_out,  // [B*S]
                       float* __restrict__ ml_part,    // [B*NCHUNK*2]
                       float* __restrict__ ctx_part) { // [B*NCHUNK*D]
    __shared__ float lds_part[GROUP * 256];
    __shared__ float lds_s[GROUP];

    const int t     = threadIdx.x;
    const int w     = t >> 5;
    const int lane  = t & 31;
    const int chunk = blockIdx.x;
    const int b     = blockIdx.y;
    const int s0    = chunk * ROWS_PER_BLOCK;

    const float4* __restrict__ keys4 = (const float4*)keys;
    const float4* __restrict__ q4    = (const float4*)query;

    const float4 qv = q4[b * (D_ / 4) + t];

    float  m = -INFINITY, l = 0.f;
    float4 ctx = {0.f, 0.f, 0.f, 0.f};

    for (int g = 0; g < NGROUPS; ++g) {
        float4 kv[GROUP];
        const size_t rowbase = ((size_t)b * S_ + (size_t)(s0 + g * GROUP)) * (D_ / 4);
        #pragma unroll
        for (int r = 0; r < GROUP; ++r) {
            kv[r] = keys4[rowbase + (size_t)r * (D_ / 4) + t];
            const float pr = kv[r].x * qv.x + kv[r].y * qv.y +
                             kv[r].z * qv.z + kv[r].w * qv.w;
            lds_part[r * 256 + t] = pr;
        }
        __syncthreads();

        // wave w reduces row w's 256 partials
        float acc = 0.f;
        #pragma unroll
        for (int i = 0; i < 8; ++i) acc += lds_part[w * 256 + lane + i * 32];
        acc = waveReduceSum(acc);
        if (lane == 0) {
            lds_s[w] = acc;
            align_out[b * S_ + s0 + g * GROUP + w] = acc;  // raw alignment score
        }
        __syncthreads();

        float sv[GROUP];
        float mnew = m;
        #pragma unroll
        for (int r = 0; r < GROUP; ++r) {
            sv[r] = lds_s[r];
            mnew  = fmaxf(mnew, sv[r]);
        }
        const float scale = expf(m - mnew);
        l     *= scale;
        ctx.x *= scale; ctx.y *= scale; ctx.z *= scale; ctx.w *= scale;
        #pragma unroll
        for (int r = 0; r < GROUP; ++r) {
            const float p = expf(sv[r] - mnew);
            l += p;
            ctx.x += p * kv[r].x;
            ctx.y += p * kv[r].y;
            ctx.z += p * kv[r].z;
            ctx.w += p * kv[r].w;
        }
        m = mnew;
        // No extra barrier needed: next-iter lds_part writes are already
        // separated from this iter's reads by the barrier above; lds_s
        // rewrites are separated by the next iter's first barrier.
    }

    const int pidx = b * NCHUNK + chunk;
    ((float4*)ctx_part)[(size_t)pidx * (D_ / 4) + t] = ctx;
    if (t == 0) {
        ml_part[pidx * 2 + 0] = m;
        ml_part[pidx * 2 + 1] = l;
    }
}

// ---------------------------------------------------------------------------
// Kernel 4: merge chunk partials, apply gauss/L, write total_context.
// grid = B, block = 256 (thread t owns d-slice [4t, 4t+4)).
// ---------------------------------------------------------------------------
__global__ __launch_bounds__(256)
void combine_kernel(const float* __restrict__ ml_part,
                    const float* __restrict__ ctx_part,
                    const float* __restrict__ g_in,
                    float* __restrict__ ctx_out) {
    const int b = blockIdx.x, t = threadIdx.x;

    float M = -INFINITY;
    for (int c = 0; c < NCHUNK; ++c)
        M = fmaxf(M, ml_part[(b * NCHUNK + c) * 2]);

    float L = 0.f;
    float4 acc = {0.f, 0.f, 0.f, 0.f};
    const float4* __restrict__ cp4 = (const float4*)ctx_part;
    for (int c = 0; c < NCHUNK; ++c) {
        const float mc = ml_part[(b * NCHUNK + c) * 2 + 0];
        const float lc = ml_part[(b * NCHUNK + c) * 2 + 1];
        const float e  = expf(mc - M);
        L += lc * e;
        const float4 cv = cp4[(size_t)(b * NCHUNK + c) * (D_ / 4) + t];
        acc.x += e * cv.x; acc.y += e * cv.y;
        acc.z += e * cv.z; acc.w += e * cv.w;
    }
    const float gs = g_in[b] / L;
    acc.x *= gs; acc.y *= gs; acc.z *= gs; acc.w *= gs;
    ((float4*)ctx_out)[b * (D_ / 4) + t] = acc;
}

// ---------------------------------------------------------------------------
extern "C" void kernel_launch(void* const* d_in, const int* in_sizes, int n_in,
                              void* d_out, int out_size, void* d_ws, size_t ws_size,
                              hipStream_t stream) {
    (void)in_sizes; (void)n_in; (void)out_size; (void)ws_size;

    const float* query       = (const float*)d_in[0];   // [B, D]
    const float* keys        = (const float*)d_in[1];   // [B, S, D]
    const float* key_lengths = (const float*)d_in[2];   // [B]
    const float* Wq          = (const float*)d_in[3];   // [D, U]
    const float* wp          = (const float*)d_in[4];   // [U]
    const int*   window_size = (const int*)d_in[5];     // scalar

    float* out       = (float*)d_out;
    float* ctx_out   = out;             // [B*D]  (total_context, first output)
    float* align_out = out + B_ * D_;   // [B*S]  (alignment_score, second output)

    // Workspace layout (floats); all regions fully written before read.
    float* ws       = (float*)d_ws;
    float* act      = ws;                          // B*U      = 32768
    float* g        = act + B_ * U_;               // B        = 32
    float* ml_part  = g + B_;                      // B*NCHUNK*2 = 2048
    float* ctx_part = ml_part + B_ * NCHUNK * 2;   // B*NCHUNK*D = 1048576

    wq_gemm_tanh_kernel<<<dim3(U_ / 16, B_ / 16), 32, 0, stream>>>(query, Wq, act);
    gauss_kernel<<<B_, 256, 0, stream>>>(act, wp, key_lengths, window_size, g);
    flash_pass_kernel<<<dim3(NCHUNK, B_), 256, 0, stream>>>(query, keys, align_out,
                                                            ml_part, ctx_part);
    combine_kernel<<<B_, 256, 0, stream>>>(ml_part, ctx_part, g, ctx_out);
}